// IndexPropagationQuantize_29789893165215
// MI455X (gfx1250) — compile-verified
//
#include <hip/hip_runtime.h>

typedef __attribute__((ext_vector_type(2))) float v2f;
typedef __attribute__((ext_vector_type(8))) float v8f;

#define N_E     8192
#define E_DIM   256
#define HWSZ    1024          // 32*32
#define ZSTRIDE_B 262144      // E_DIM*HWSZ (per-batch stride in floats)
#define BSTRIDE 260           // padded LDS row stride for W tile (floats)
#define NT      (N_E / 16)    // 512 N-tiles

// ---------------------------------------------------------------------------
// CDNA5 async global->LDS copy (ASYNCcnt-tracked, no VGPR landing zone).
// VDST operand = wave-relative LDS byte offset = low 32 bits of flat address
// (ISA 10.2: LDS aperture uses the address's low 32 bits).
// ---------------------------------------------------------------------------
__device__ __forceinline__
void async_b128_to_lds(const float* gptr, const float* lptr) {
    uint32_t lds_off = (uint32_t)(uintptr_t)lptr;
    uint64_t gaddr   = (uint64_t)(uintptr_t)gptr;
    asm volatile("global_load_async_to_lds_b128 %0, %1, off"
                 :: "v"(lds_off), "v"(gaddr) : "memory");
}

// Stage one 16x256 f32 codebook tile into LDS (padded rows), 64 threads,
// 16 async b128 per lane-thread -> ASYNCcnt += 16 per wave.
__device__ __forceinline__
void stage_tile(const float* __restrict__ Wt, float* dst, int tid) {
    #pragma unroll
    for (int j = 0; j < 16; ++j) {
        int s  = tid + j * 64;           // float4 slot, 1024 total
        int n  = s >> 6;
        int d4 = (s & 63) * 4;
        async_b128_to_lds(Wt + n * E_DIM + d4, dst + n * BSTRIDE + d4);
    }
}

// ---------------------------------------------------------------------------
// Kernel 0: zero the loss accumulator (d_ws is poisoned 0xAA; must re-zero).
// ---------------------------------------------------------------------------
__global__ void vq_zero_acc(float* acc) {
    if (threadIdx.x == 0) acc[0] = 0.0f;
}

// ---------------------------------------------------------------------------
// Kernel 1: logits GEMM (M=8192 pixels, N=8192 codes, K=256) fused with
// row-argmax. A fragments live in VGPRs (reused over all 512 N-tiles);
// B tiles are double-buffered in LDS via async copies overlapped with
// v_wmma_f32_16x16x4_f32 compute. Block = 64 threads (2 waves), each wave
// owns one 16-pixel M-tile.
// ---------------------------------------------------------------------------
__global__ __launch_bounds__(64)
void vq_argmax_wmma(const float* __restrict__ z,
                    const float* __restrict__ Wm,
                    int* __restrict__ wsInd,
                    float* __restrict__ outInd) {
    __shared__ float ldsB[2][16 * BSTRIDE];      // double-buffered W tile

    const int tid   = threadIdx.x;
    const int wave  = tid >> 5;
    const int lane  = tid & 31;
    const int m     = lane & 15;
    const int koff  = (lane >> 4) << 1;          // 0 for lanes 0-15, 2 for 16-31
    const int mtile = blockIdx.x * 2 + wave;
    const int p0    = mtile * 16;                // first pixel of this M-tile
    const int bImg  = p0 >> 10;                  // batch index (tiles never straddle)
    const int hw0   = p0 & (HWSZ - 1);

    // --- A fragments in registers: lane holds z[bImg, k0/k0+1, hw0+m] per chunk.
    v2f areg[64];
    {
        const float* zb = z + (size_t)bImg * ZSTRIDE_B + hw0 + m;
        #pragma unroll
        for (int c = 0; c < 64; ++c) {
            int k0 = 4 * c + koff;
            areg[c].x = zb[(size_t)k0 * HWSZ];
            areg[c].y = zb[(size_t)(k0 + 1) * HWSZ];
        }
    }

    float best[8];
    int   bidx[8];
    #pragma unroll
    for (int v = 0; v < 8; ++v) { best[v] = -3.0e38f; bidx[v] = 0; }

    // Prime buffer 0.
    stage_tile(Wm, &ldsB[0][0], tid);

    for (int nt = 0; nt < NT; ++nt) {
        const int cur = nt & 1;
        __syncthreads();                         // prior reads of buf cur^1 done
        if (nt + 1 < NT) {
            stage_tile(Wm + (size_t)(nt + 1) * 16 * E_DIM, &ldsB[cur ^ 1][0], tid);
            // async loads complete in order: <=16 outstanding => buf cur landed
            asm volatile("s_wait_asynccnt 0x10" ::: "memory");
        } else {
            asm volatile("s_wait_asynccnt 0x0" ::: "memory");
        }
        __syncthreads();                         // tile visible to both waves

        // --- 64 chained f32 WMMAs: C(16x16) += A(16x4) * B(4x16)
        const float* pB = &ldsB[cur][(size_t)m * BSTRIDE];
        v8f acc = {};
        #pragma unroll
        for (int c = 0; c < 64; ++c) {
            int k0 = 4 * c + koff;
            v2f bv = { pB[k0], pB[k0 + 1] };     // adjacent -> one ds_load_b64
            acc = __builtin_amdgcn_wmma_f32_16x16x4_f32(
                false, areg[c], false, bv, (short)0, acc, false, false);
        }

        // --- Fold C tile into running argmax.
        // C layout: VGPR v, lanes 0-15 -> (M=v, N=lane); lanes 16-31 -> (M=v+8, N=lane-16)
        const int ncol = nt * 16 + m;
        #pragma unroll
        for (int v = 0; v < 8; ++v) {
            float val = acc[v];
            if (val > best[v]) { best[v] = val; bidx[v] = ncol; }
        }
    }

    // --- Cross-lane argmax over the 16 lanes sharing each M (xor stays in half).
    #pragma unroll
    for (int v = 0; v < 8; ++v) {
        #pragma unroll
        for (int off = 8; off >= 1; off >>= 1) {
            float ov = __shfl_xor(best[v], off, 32);
            int   oi = __shfl_xor(bidx[v], off, 32);
            if (ov > best[v] || (ov == best[v] && oi < bidx[v])) {
                best[v] = ov; bidx[v] = oi;
            }
        }
    }

    if (m == 0) {
        int half = lane >> 4;                    // lane 0 -> M=v, lane 16 -> M=v+8
        #pragma unroll
        for (int v = 0; v < 8; ++v) {
            int pix = p0 + v + half * 8;
            wsInd[pix]  = bidx[v];
            outInd[pix] = (float)bidx[v];
        }
    }
}

// ---------------------------------------------------------------------------
// Kernel 2: z_q = W[ind] scattered to (b,d,h,w) layout + sum (W[ind]-z)^2.
// One block per (b,d) row -> coalesced z/z_q along hw; W gather hits L2.
// ---------------------------------------------------------------------------
__global__ __launch_bounds__(256)
void vq_gather_loss(const float* __restrict__ z,
                    const float* __restrict__ Wm,
                    const int* __restrict__ wsInd,
                    float* __restrict__ zq,
                    float* __restrict__ acc) {
    const int bd = blockIdx.x;                   // 0..2047
    const int b  = bd >> 8;
    const int d  = bd & 255;
    const size_t base = (size_t)b * ZSTRIDE_B + (size_t)d * HWSZ;

    float sq = 0.0f;
    #pragma unroll
    for (int j = 0; j < 4; ++j) {
        int hw  = threadIdx.x + j * 256;
        int p   = (b << 10) + hw;
        int idx = wsInd[p];
        float wv = Wm[(size_t)idx * E_DIM + d];
        float zv = z[base + hw];
        zq[base + hw] = wv;
        float df = wv - zv;
        sq += df * df;
    }
    #pragma unroll
    for (int off = 16; off >= 1; off >>= 1)
        sq += __shfl_xor(sq, off, 32);
    if ((threadIdx.x & 31) == 0) atomicAdd(acc, sq);
}

// ---------------------------------------------------------------------------
// Kernel 3: quant_loss = 2.25 * sum / 2097152
// ---------------------------------------------------------------------------
__global__ void vq_finalize(const float* acc, float* qloss) {
    if (threadIdx.x == 0)
        qloss[0] = 2.25f * acc[0] * (1.0f / 2097152.0f);
}

extern "C" void kernel_launch(void* const* d_in, const int* in_sizes, int n_in,
                              void* d_out, int out_size, void* d_ws, size_t ws_size,
                              hipStream_t stream) {
    const float* z  = (const float*)d_in[0];     // (8,256,32,32)
    const float* Wm = (const float*)d_in[1];     // (8192,256)

    float* out    = (float*)d_out;
    float* zq     = out;                          // 2097152 floats
    float* qloss  = out + 2097152;                // 1 float
    float* outInd = out + 2097153;                // 8192 floats (indices as f32)

    int*   wsInd  = (int*)d_ws;                   // 8192 ints
    float* acc    = (float*)((char*)d_ws + 32768);

    vq_zero_acc   <<<1,    1,   0, stream>>>(acc);
    vq_argmax_wmma<<<256,  64,  0, stream>>>(z, Wm, wsInd, outInd);
    vq_gather_loss<<<2048, 256, 0, stream>>>(z, Wm, wsInd, zq, acc);
    vq_finalize   <<<1,    1,   0, stream>>>(acc, qloss);
}